// Model_89069031785173
// MI455X (gfx1250) — compile-verified
//
#include <hip/hip_runtime.h>
#include <math.h>

typedef __attribute__((ext_vector_type(2))) float v2f;
typedef __attribute__((ext_vector_type(8))) float v8f;

#define DEPTH 9
#define NCLS  40
#define MT    4   // row tiles (of 16) per wave

// ---------------------------------------------------------------------------
// Weight repack: w[k][c][n] -> wp[((k*(Cin/4)+c/4)*Cout + n)*4 + (c&3)]
// so a lane's B pair (K = c0+2*half, c0+2*half+1) is one contiguous b64 load.
// ---------------------------------------------------------------------------
__global__ void repack_w(const float* __restrict__ w, float* __restrict__ wp,
                         int K, int Cin, int Cout)
{
    size_t total = (size_t)K * Cin * Cout;
    int cc4 = Cin >> 2;
    for (size_t i = (size_t)blockIdx.x * blockDim.x + threadIdx.x; i < total;
         i += (size_t)gridDim.x * blockDim.x) {
        int n   = (int)(i % (size_t)Cout);
        size_t t = i / (size_t)Cout;
        int c   = (int)(t % (size_t)Cin);
        int k   = (int)(t / (size_t)Cin);
        wp[(((size_t)k * cc4 + (c >> 2)) * Cout + n) * 4 + (c & 3)] = w[i];
    }
}

// ---------------------------------------------------------------------------
// Rulebook sparse-conv via fp32 WMMA 16x16x4.
//   out[r, c] = sum_k  xp[ idx[k][r] ] @ w[k]      (+ bias[c] if bias)
// xp has Nin+1 rows, row Nin all zeros (pad row; idx pad value == Nin).
// idx == nullptr -> identity rulebook (dense NIN matmul), K == 1.
// One wave computes a 64x16 output tile: 4 row tiles share each B fragment.
// A (16x4 f32): lanes 0-15 = rows, v0=K0 v1=K1; lanes 16-31: v0=K2 v1=K3.
// B (4x16 f32): lanes = columns;   v0/v1 = K rows (0,1 | 2,3 per half).
// C/D (16x16 f32): lanes = columns; VGPR v = row v (lanes 0-15) / v+8.
// ---------------------------------------------------------------------------
__global__ __launch_bounds__(32) void sconv_wmma(
    const float* __restrict__ xp, const int* __restrict__ idx,
    const float* __restrict__ wp,   // repacked weights
    const float* __restrict__ bias,
    float* __restrict__ out,
    int Nout, int Nin, int Cin, int Cout, int K)
{
    const int lane = threadIdx.x;
    const int half = lane >> 4;          // 0: K0/K1, 1: K2/K3
    const int l15  = lane & 15;
    const int tileM0 = blockIdx.x * (16 * MT);
    const int tileN  = blockIdx.y * 16;

    const int col    = tileN + l15;      // output column this lane owns
    const bool colOK = (col < Cout);
    const int  colC  = colOK ? col : (Cout - 1);
    const int  cc4   = Cin >> 2;

    v8f acc[MT];
#pragma unroll
    for (int t = 0; t < MT; ++t) acc[t] = (v8f){};

    for (int k = 0; k < K; ++k) {
        // gathered input rows for the 4 row tiles
        const float* xr[MT];
#pragma unroll
        for (int t = 0; t < MT; ++t) {
            int arow = tileM0 + t * 16 + l15;
            int row;
            if (idx) {
                row = (arow < Nout) ? idx[(size_t)k * Nout + arow] : Nin;
            } else {
                row = (arow < Nout) ? arow : Nin;
            }
            xr[t] = xp + (size_t)row * Cin + 2 * half;
        }
        // repacked B base for this k / this lane's column / this lane's K pair
        const float* __restrict__ wb =
            wp + ((size_t)k * cc4 * Cout + colC) * 4 + 2 * half;

        for (int c0 = 0; c0 < Cin; c0 += 4) {
            const float* __restrict__ wq = wb + (size_t)(c0 >> 2) * Cout * 4;
            v2f B;
            B[0] = wq[0];
            B[1] = wq[1];
            B[0] = colOK ? B[0] : 0.0f;  // v_cndmask; EXEC stays all-ones
            B[1] = colOK ? B[1] : 0.0f;
#pragma unroll
            for (int t = 0; t < MT; ++t) {
                v2f A;
                A[0] = xr[t][c0];
                A[1] = xr[t][c0 + 1];
                acc[t] = __builtin_amdgcn_wmma_f32_16x16x4_f32(
                    /*neg_a=*/false, A, /*neg_b=*/false, B,
                    /*c_mod=*/(short)0, acc[t],
                    /*reuse_a=*/false, /*reuse_b=*/false);
            }
        }
    }

    float bval = 0.0f;
    if (bias && colOK) bval = bias[col];
#pragma unroll
    for (int t = 0; t < MT; ++t) {
#pragma unroll
        for (int v = 0; v < 8; ++v) {
            int r = tileM0 + t * 16 + v + 8 * half;
            if (r < Nout && colOK)
                out[(size_t)r * Cout + col] = acc[t][v] + bval;
        }
    }
}

// ---------------------------------------------------------------------------
// Per-channel batch stats -> fused affine: a = s*rsqrt(var+eps), b = bias-mu*a
// One block per channel; ab holds [a(0..C-1), b(0..C-1)].
// ---------------------------------------------------------------------------
__global__ __launch_bounds__(256) void bnstats(
    const float* __restrict__ x, int N, int C,
    const float* __restrict__ scale, const float* __restrict__ bias,
    float* __restrict__ ab)
{
    __shared__ float ssum[256];
    __shared__ float ssq[256];
    const int c = blockIdx.x;
    float s = 0.0f, q = 0.0f;
    for (int r = threadIdx.x; r < N; r += 256) {
        float v = x[(size_t)r * C + c];
        s += v;
        q += v * v;
    }
    ssum[threadIdx.x] = s;
    ssq[threadIdx.x]  = q;
    __syncthreads();
    for (int off = 128; off > 0; off >>= 1) {
        if (threadIdx.x < off) {
            ssum[threadIdx.x] += ssum[threadIdx.x + off];
            ssq[threadIdx.x]  += ssq[threadIdx.x + off];
        }
        __syncthreads();
    }
    if (threadIdx.x == 0) {
        float invN = 1.0f / (float)N;
        float mu   = ssum[0] * invN;
        float var  = ssq[0] * invN - mu * mu;
        float a    = scale[c] * rsqrtf(var + 1e-4f);
        ab[c]      = a;
        ab[C + c]  = bias[c] - mu * a;
    }
}

// y = relu(a*x + b), writes N+1 rows (pad row = 0)
__global__ void bnrelu(const float* __restrict__ x, const float* __restrict__ ab,
                       float* __restrict__ y, int N, int C)
{
    size_t total = (size_t)(N + 1) * (size_t)C;
    for (size_t i = (size_t)blockIdx.x * blockDim.x + threadIdx.x; i < total;
         i += (size_t)gridDim.x * blockDim.x) {
        int c = (int)(i % (size_t)C);
        size_t r = i / (size_t)C;
        float v = 0.0f;
        if (r < (size_t)N) v = fmaxf(fmaf(x[i], ab[c], ab[C + c]), 0.0f);
        y[i] = v;
    }
}

// dst = src with an extra zeroed pad row appended
__global__ void pad_copy(const float* __restrict__ src, float* __restrict__ dst,
                         int N, int C)
{
    size_t total = (size_t)(N + 1) * (size_t)C;
    size_t nc    = (size_t)N * (size_t)C;
    for (size_t i = (size_t)blockIdx.x * blockDim.x + threadIdx.x; i < total;
         i += (size_t)gridDim.x * blockDim.x) {
        dst[i] = (i < nc) ? src[i] : 0.0f;
    }
}

// dst[r,c] = skip[r,c] + ynext[parent[r], c]   (dst may alias skip)
__global__ void unpool_add(const float* __restrict__ skip,
                           const float* __restrict__ ynext,
                           const int* __restrict__ parent,
                           float* __restrict__ dst, int N, int C)
{
    size_t total = (size_t)N * (size_t)C;
    for (size_t i = (size_t)blockIdx.x * blockDim.x + threadIdx.x; i < total;
         i += (size_t)gridDim.x * blockDim.x) {
        int c = (int)(i % (size_t)C);
        size_t r = i / (size_t)C;
        dst[i] = skip[i] + ynext[(size_t)parent[r] * C + c];
    }
}

static inline unsigned div_up(size_t n, size_t d) { return (unsigned)((n + d - 1) / d); }

extern "C" void kernel_launch(void* const* d_in, const int* in_sizes, int n_in,
                              void* d_out, int out_size, void* d_ws, size_t ws_size,
                              hipStream_t stream)
{
    (void)n_in; (void)out_size; (void)ws_size;
    int Cc[DEPTH];
    for (int i = 0; i < DEPTH; ++i) Cc[i] = 16 * (i + 1);

    const float* feats = (const float*)d_in[0];
    const float* w_in  = (const float*)d_in[2];
    const float* sub_w[DEPTH];   for (int i = 0; i < DEPTH; ++i) sub_w[i]  = (const float*)d_in[3 + i];
    const float* nin_w[DEPTH];   for (int i = 0; i < DEPTH; ++i) nin_w[i]  = (const float*)d_in[12 + i];
    const float* nin_b = (const float*)d_in[21];
    const float* down_w[DEPTH-1];for (int i = 0; i < DEPTH-1; ++i) down_w[i] = (const float*)d_in[22 + i];
    const float* bnfs[DEPTH];    for (int i = 0; i < DEPTH; ++i) bnfs[i]   = (const float*)d_in[30 + i];
    const float* bnfb[DEPTH];    for (int i = 0; i < DEPTH; ++i) bnfb[i]   = (const float*)d_in[39 + i];
    const float* bnbs[DEPTH];    for (int i = 0; i < DEPTH; ++i) bnbs[i]   = (const float*)d_in[48 + i];
    const float* bnbb[DEPTH];    for (int i = 0; i < DEPTH; ++i) bnbb[i]   = (const float*)d_in[57 + i];
    const float* bnds[DEPTH-1];  for (int i = 0; i < DEPTH-1; ++i) bnds[i] = (const float*)d_in[66 + i];
    const float* bndb[DEPTH-1];  for (int i = 0; i < DEPTH-1; ++i) bndb[i] = (const float*)d_in[74 + i];
    const int* sub_idx[DEPTH];   for (int i = 0; i < DEPTH; ++i) sub_idx[i]  = (const int*)d_in[82 + i];
    const int* down_idx[DEPTH-1];for (int i = 0; i < DEPTH-1; ++i) down_idx[i] = (const int*)d_in[91 + i];
    const int* parent_idx[DEPTH-1];for (int i = 0; i < DEPTH-1; ++i) parent_idx[i] = (const int*)d_in[99 + i];

    int Nlev[DEPTH];
    for (int i = 0; i < DEPTH; ++i) Nlev[i] = in_sizes[82 + i] / 27;

    // --- carve workspace ---
    char* wsb = (char*)d_ws;
    size_t off = 0;
    auto alloc = [&](size_t nfloats) -> float* {
        float* p = (float*)(wsb + off);
        off += ((nfloats * sizeof(float)) + 255) & ~(size_t)255;
        return p;
    };
    float* featsP = alloc((size_t)(Nlev[0] + 1) * 4);
    float* xbuf[DEPTH];
    size_t maxNC = 0;
    for (int i = 0; i < DEPTH; ++i) {
        size_t sz = (size_t)(Nlev[i] + 1) * Cc[i];
        xbuf[i] = alloc(sz);
        if (sz > maxNC) maxNC = sz;
    }
    float* hbuf = alloc(maxNC);
    float* ybuf[DEPTH];
    for (int i = 0; i < DEPTH; ++i) ybuf[i] = alloc((size_t)Nlev[i] * NCLS);
    float* ab = alloc(2 * 144);
    float* wpbuf = alloc((size_t)27 * 144 * 144);   // max repacked weight

    const int N0 = Nlev[0];
    const unsigned RB = 16 * MT;   // rows per conv block

    auto launch_conv = [&](const float* x, const int* idx, const float* w,
                           const float* bias, float* o,
                           int No, int Ni, int Ci, int Co, int K) {
        size_t we = (size_t)K * Ci * Co;
        repack_w<<<div_up(we, 256), 256, 0, stream>>>(w, wpbuf, K, Ci, Co);
        sconv_wmma<<<dim3(div_up((size_t)No, RB), div_up((size_t)Co, 16)),
                     32, 0, stream>>>(x, idx, wpbuf, bias, o, No, Ni, Ci, Co, K);
    };

    // feats -> padded
    {
        size_t tot = (size_t)(N0 + 1) * 4;
        pad_copy<<<div_up(tot, 256), 256, 0, stream>>>(feats, featsP, N0, 4);
    }
    // input conv: [N0,4] -> [N0,16], 27-point rulebook
    launch_conv(featsP, sub_idx[0], w_in, nullptr, xbuf[0], N0, N0, 4, 16, 27);

    for (int i = 0; i < DEPTH; ++i) {
        const int N = Nlev[i], C = Cc[i];
        const size_t totP = (size_t)(N + 1) * C;

        // foo: BNReLU -> submanifold 3x3x3 conv
        bnstats<<<C, 256, 0, stream>>>(xbuf[i], N, C, bnfs[i], bnfb[i], ab);
        bnrelu<<<div_up(totP, 256), 256, 0, stream>>>(xbuf[i], ab, hbuf, N, C);
        launch_conv(hbuf, sub_idx[i], sub_w[i], nullptr, xbuf[i], N, N, C, C, 27);

        // bar: BNReLU -> NIN (bias only at deepest level)
        bnstats<<<C, 256, 0, stream>>>(xbuf[i], N, C, bnbs[i], bnbb[i], ab);
        bnrelu<<<div_up(totP, 256), 256, 0, stream>>>(xbuf[i], ab, hbuf, N, C);
        launch_conv(hbuf, nullptr, nin_w[i], (i == DEPTH - 1) ? nin_b : nullptr,
                    ybuf[i], N, N, C, NCLS, 1);

        // down: BNReLU -> 2x2x2 stride-2 conv
        if (i < DEPTH - 1) {
            const int Nn = Nlev[i + 1], Cn = Cc[i + 1];
            bnstats<<<C, 256, 0, stream>>>(xbuf[i], N, C, bnds[i], bndb[i], ab);
            bnrelu<<<div_up(totP, 256), 256, 0, stream>>>(xbuf[i], ab, hbuf, N, C);
            launch_conv(hbuf, down_idx[i], down_w[i], nullptr, xbuf[i + 1],
                        Nn, N, C, Cn, 8);
        }
    }

    // decoder: y_i = skip_i + y_{i+1}[parent_idx_i]
    for (int i = DEPTH - 2; i >= 0; --i) {
        float* dst = (i == 0) ? (float*)d_out : ybuf[i];
        size_t tot = (size_t)Nlev[i] * NCLS;
        unpool_add<<<div_up(tot, 256), 256, 0, stream>>>(
            ybuf[i], ybuf[i + 1], parent_idx[i], dst, Nlev[i], NCLS);
    }
}